// LinearAttention_75557064671401
// MI455X (gfx1250) — compile-verified
//
#include <hip/hip_runtime.h>
#include <hip/hip_bf16.h>
#include <math.h>
#include <stdint.h>

typedef float v2f __attribute__((ext_vector_type(2)));
typedef float v8f __attribute__((ext_vector_type(8)));

#define BATCH 4
#define FDIM 1024
#define IDIM 2048
#define SDIM 2048
#define KW 7
#define GRP 4
#define CPG 512            // IDIM / GRP

// ---------------------------------------------------------------------------
// CDNA5 async global->LDS DMA staging (tracked by ASYNCcnt).
// lds_off: byte offset within LDS (low 32 bits of generic shared address).
// base:    SGPR-uniform global base pointer; off: per-lane byte offset.
// ---------------------------------------------------------------------------
__device__ __forceinline__ void async_copy_b128(uint32_t lds_off,
                                                const float* base,
                                                uint32_t off) {
  asm volatile("global_load_async_to_lds_b128 %0, %1, %2"
               :
               : "v"(lds_off), "v"(off), "s"(base)
               : "memory");
}

__device__ __forceinline__ void wait_async() {
#if defined(__has_builtin)
#if __has_builtin(__builtin_amdgcn_s_wait_asynccnt)
  __builtin_amdgcn_s_wait_asynccnt(0);
  return;
#endif
#endif
  asm volatile("s_wait_asynccnt 0" ::: "memory");
}

// ---------------------------------------------------------------------------
// fp32 WMMA GEMM:  C[b] = A[M,Kd] x B[b][Kd,N]   (A shared over batch)
// Block tile 128(M) x 128(N), 8 waves (2 M x 4 N), wave tile 64x32 =
// 4x2 WMMA 16x16 tiles -> 8 wmma per 6 fragment loads per k-step.
// Operands staged via async DMA to LDS, K chunked by 16.
// ---------------------------------------------------------------------------
__global__ __launch_bounds__(256) void gemm_f32_wmma(
    const float* __restrict__ A,   // [M, Kd] row-major
    const float* __restrict__ Bm,  // [batch, Kd, N] row-major
    float* __restrict__ C,         // [batch, M, N]
    int M, int Kd, int N)
{
  __shared__ float As[128][18];    // 16 + pad2
  __shared__ float Bs[16][132];    // 128 + pad4

  const int b  = blockIdx.z;
  const int m0 = blockIdx.y * 128;
  const int n0 = blockIdx.x * 128;
  const float* Bp = Bm + (size_t)b * Kd * N;
  float* Cp = C + (size_t)b * M * N;

  const int tid  = threadIdx.x;
  const int lane = tid & 31;
  const int wave = tid >> 5;
  const int wm = (wave & 1) * 64;   // 2 waves along M
  const int wn = (wave >> 1) * 32;  // 4 waves along N

  v8f acc[4][2];
#pragma unroll
  for (int i = 0; i < 4; ++i)
#pragma unroll
    for (int j = 0; j < 2; ++j) acc[i][j] = {};

  // staging geometry (per-thread, loop-invariant)
  const int am  = tid >> 1;            // A row, 8 floats each
  const int ak8 = (tid & 1) * 8;
  const uint32_t lds_a = (uint32_t)(uintptr_t)&As[am][ak8];
  const int bk  = tid >> 4;            // B row, 8 floats each
  const int bn8 = (tid & 15) * 8;
  const uint32_t lds_b = (uint32_t)(uintptr_t)&Bs[bk][bn8];

  for (int kk = 0; kk < Kd; kk += 16) {
    // async DMA stage A tile 128x16 and B tile 16x128
    uint32_t aoff = (uint32_t)((((size_t)(m0 + am)) * Kd + kk + ak8) * 4);
    async_copy_b128(lds_a,      A, aoff);
    async_copy_b128(lds_a + 16, A, aoff + 16);
    uint32_t boff = (uint32_t)(((size_t)(kk + bk) * N + n0 + bn8) * 4);
    async_copy_b128(lds_b,      Bp, boff);
    async_copy_b128(lds_b + 16, Bp, boff + 16);
    wait_async();
    __syncthreads();

#pragma unroll
    for (int ks = 0; ks < 16; ks += 4) {
      const int ka = ks + 2 * (lane >> 4);
      v2f afr[4], bfr[2];
#pragma unroll
      for (int i = 0; i < 4; ++i) {
        int m = wm + i * 16 + (lane & 15);
        afr[i].x = As[m][ka];
        afr[i].y = As[m][ka + 1];
      }
#pragma unroll
      for (int j = 0; j < 2; ++j) {
        int n = wn + j * 16 + (lane & 15);
        bfr[j].x = Bs[ka][n];
        bfr[j].y = Bs[ka + 1][n];
      }
#pragma unroll
      for (int i = 0; i < 4; ++i)
#pragma unroll
        for (int j = 0; j < 2; ++j)
          acc[i][j] = __builtin_amdgcn_wmma_f32_16x16x4_f32(
              false, afr[i], false, bfr[j], (short)0, acc[i][j], false, false);
    }
    __syncthreads();
  }

  // C/D layout: VGPR r -> M = r + 8*(lane>>4), N = lane&15
#pragma unroll
  for (int i = 0; i < 4; ++i)
#pragma unroll
    for (int j = 0; j < 2; ++j) {
      int mb = m0 + wm + i * 16 + 8 * (lane >> 4);
      int nb = n0 + wn + j * 16 + (lane & 15);
#pragma unroll
      for (int r = 0; r < 8; ++r)
        Cp[(size_t)(mb + r) * N + nb] = acc[i][j][r];
    }
}

// ---------------------------------------------------------------------------
// Grouped causal conv1d as WMMA GEMM, accumulating over c (512) and tap k (7).
// B tile [16c x 134s] staged once per c-chunk (regular loads: causal zero
// fill) and reused by all 7 taps via shifted reads; A tile [128o x 16c]
// async-DMA'd per tap from transposed weights.
// ---------------------------------------------------------------------------
__global__ __launch_bounds__(256) void gconv_f32_wmma(
    const float* __restrict__ Wt,  // [7][6144][512]  (k, o, c)
    const float* __restrict__ X,   // [batch][2048][S]
    float* __restrict__ Y,         // [batch][6144][S]
    int S)
{
  __shared__ float As[128][18];
  __shared__ float Bs[16][136];    // 134 + pad2

  const int b     = blockIdx.z;
  const int mt    = blockIdx.y;      // 0..47
  const int group = mt / 12;         // 1536/128 = 12 tiles per group
  const int m0    = mt * 128;        // global out-channel base
  const int cg0   = group * CPG;     // input-channel base for this group
  const int n0    = blockIdx.x * 128;

  const int tid  = threadIdx.x;
  const int lane = tid & 31;
  const int wave = tid >> 5;
  const int wm = (wave & 1) * 64;
  const int wn = (wave >> 1) * 32;

  v8f acc[4][2];
#pragma unroll
  for (int i = 0; i < 4; ++i)
#pragma unroll
    for (int j = 0; j < 2; ++j) acc[i][j] = {};

  const int am  = tid >> 1;
  const int ak8 = (tid & 1) * 8;
  const uint32_t lds_a = (uint32_t)(uintptr_t)&As[am][ak8];

  for (int c0 = 0; c0 < CPG; c0 += 16) {
    __syncthreads();  // previous readers of Bs/As done
    // stage B tile: X[b][cg0+c0+c][n0-6+t], t in [0,134), zero-fill s<0
    for (int idx = tid; idx < 16 * 134; idx += 256) {
      int c = idx / 134, t = idx % 134;
      int s = n0 - 6 + t;
      float v = (s >= 0)
          ? X[((size_t)b * IDIM + cg0 + c0 + c) * S + s] : 0.0f;
      Bs[c][t] = v;
    }
    for (int k = 0; k < KW; ++k) {
      if (k > 0) __syncthreads();  // previous As readers done
      uint32_t aoff = (uint32_t)((((size_t)k * 6144 + m0 + am) * CPG
                                  + c0 + ak8) * 4);
      async_copy_b128(lds_a,      Wt, aoff);
      async_copy_b128(lds_a + 16, Wt, aoff + 16);
      wait_async();
      __syncthreads();  // As (and, at k==0, Bs) visible
#pragma unroll
      for (int ks = 0; ks < 16; ks += 4) {
        const int ka = ks + 2 * (lane >> 4);
        v2f afr[4], bfr[2];
#pragma unroll
        for (int i = 0; i < 4; ++i) {
          int m = wm + i * 16 + (lane & 15);
          afr[i].x = As[m][ka];
          afr[i].y = As[m][ka + 1];
        }
#pragma unroll
        for (int j = 0; j < 2; ++j) {
          int n = wn + j * 16 + (lane & 15) + k;  // shifted tap read
          bfr[j].x = Bs[ka][n];
          bfr[j].y = Bs[ka + 1][n];
        }
#pragma unroll
        for (int i = 0; i < 4; ++i)
#pragma unroll
          for (int j = 0; j < 2; ++j)
            acc[i][j] = __builtin_amdgcn_wmma_f32_16x16x4_f32(
                false, afr[i], false, bfr[j], (short)0, acc[i][j], false, false);
      }
    }
  }

#pragma unroll
  for (int i = 0; i < 4; ++i)
#pragma unroll
    for (int j = 0; j < 2; ++j) {
      int mb = m0 + wm + i * 16 + 8 * (lane >> 4);
      int nb = n0 + wn + j * 16 + (lane & 15);
#pragma unroll
      for (int r = 0; r < 8; ++r)
        Y[((size_t)b * 6144 + mb + r) * S + nb] = acc[i][j][r];
    }
}

// ---------------------------------------------------------------------------
// Wave-parallel inclusive cumsum along S per (b,i) row, fused with
// t = cum/divisor*scale + shift.  One wave32 per row.
// ---------------------------------------------------------------------------
__global__ __launch_bounds__(256) void scan_kernel(
    const float* __restrict__ X1,      // [batch][6144][S]
    const float* __restrict__ divisor, // [S]
    float* __restrict__ T,             // [batch][2048][S]
    int S)
{
  int gw   = (blockIdx.x * blockDim.x + threadIdx.x) >> 5;
  int lane = threadIdx.x & 31;
  int b = gw / IDIM;
  int i = gw % IDIM;
  const float* depth = X1 + ((size_t)b * 6144 + i) * S;
  const float* scale = X1 + ((size_t)b * 6144 + IDIM + i) * S;
  const float* shift = X1 + ((size_t)b * 6144 + 2 * IDIM + i) * S;
  float* t = T + ((size_t)b * IDIM + i) * S;

  float carry = 0.0f;
  for (int s0 = 0; s0 < S; s0 += 32) {
    int s = s0 + lane;
    float v = depth[s];
#pragma unroll
    for (int d = 1; d < 32; d <<= 1) {
      float u = __shfl_up(v, d, 32);
      if (lane >= d) v += u;
    }
    v += carry;
    carry = __shfl(v, 31, 32);
    t[s] = v / divisor[s] * scale[s] + shift[s];
  }
}

// ---------------------------------------------------------------------------
// Channel mean/RMS norm + leaky-relu over I=2048 channels per (b,s).
// FUSE=true: input is conv output [b][6144][S], u = s0*s1 + sh on the fly.
// Block: 32 s-positions x 8 reduction rows.
// ---------------------------------------------------------------------------
template <bool FUSE>
__global__ __launch_bounds__(256) void norm_kernel(
    const float* __restrict__ U,
    float* __restrict__ Yn,   // [batch][2048][S]
    int S)
{
  __shared__ float red1[8][33];
  __shared__ float red2[8][33];
  __shared__ float smean[32];
  __shared__ float srstd[32];

  const int b  = blockIdx.y;
  const int tx = threadIdx.x & 31;
  const int ty = threadIdx.x >> 5;
  const int s  = blockIdx.x * 32 + tx;
  const size_t base = FUSE ? ((size_t)b * 6144 * S) : ((size_t)b * IDIM * S);

  float sum = 0.0f, sq = 0.0f;
  for (int i = ty; i < IDIM; i += 8) {
    float u;
    if (FUSE) {
      float a0 = U[base + (size_t)i * S + s];
      float a1 = U[base + (size_t)(i + IDIM) * S + s];
      float a2 = U[base + (size_t)(i + 2 * IDIM) * S + s];
      u = a0 * a1 + a2;
    } else {
      u = U[base + (size_t)i * S + s];
    }
    sum += u; sq += u * u;
  }
  red1[ty][tx] = sum;
  red2[ty][tx] = sq;
  __syncthreads();
  if (ty == 0) {
    float tsum = 0.0f, tsq = 0.0f;
#pragma unroll
    for (int r = 0; r < 8; ++r) { tsum += red1[r][tx]; tsq += red2[r][tx]; }
    float mean = tsum * (1.0f / (float)IDIM);
    float var  = tsq * (1.0f / (float)IDIM) - mean * mean;
    var = var > 0.0f ? var : 0.0f;
    smean[tx] = mean;
    srstd[tx] = 1.0f / (sqrtf(var) + 1e-5f);
  }
  __syncthreads();
  const float mean = smean[tx];
  const float rstd = srstd[tx];
  for (int i = ty; i < IDIM; i += 8) {
    float u;
    if (FUSE) {
      float a0 = U[base + (size_t)i * S + s];
      float a1 = U[base + (size_t)(i + IDIM) * S + s];
      float a2 = U[base + (size_t)(i + 2 * IDIM) * S + s];
      u = a0 * a1 + a2;
    } else {
      u = U[base + (size_t)i * S + s];
    }
    float y = (u - mean) * rstd;
    y = y >= 0.0f ? y : 0.02f * y;
    Yn[((size_t)b * IDIM + i) * S + s] = y;
  }
}

// ---------------------------------------------------------------------------
// w1 [o][c][k] -> w1t [k][o][c] so conv A-tiles load coalesced.
// ---------------------------------------------------------------------------
__global__ void w1_transpose(const float* __restrict__ w1,
                             float* __restrict__ w1t, int total)
{
  int idx = blockIdx.x * blockDim.x + threadIdx.x;
  if (idx >= total) return;
  int k = idx % KW;
  int c = (idx / KW) % CPG;
  int o = idx / (KW * CPG);
  w1t[((size_t)k * 6144 + o) * CPG + c] = w1[idx];
}

extern "C" void kernel_launch(void* const* d_in, const int* in_sizes, int n_in,
                              void* d_out, int out_size, void* d_ws, size_t ws_size,
                              hipStream_t stream) {
  const float* inp     = (const float*)d_in[0];  // [4,1024,2048]
  const float* divisor = (const float*)d_in[1];  // [2048]
  const float* w0      = (const float*)d_in[2];  // [6144,1024]
  const float* w1      = (const float*)d_in[3];  // [6144,512,7]
  const float* w2      = (const float*)d_in[4];  // [1024,2048]
  float* out = (float*)d_out;                    // [4,1024,2048]

  char* ws = (char*)d_ws;
  size_t off = 0;
  float* x1  = (float*)(ws + off); off += (size_t)BATCH * 6144 * SDIM * 4; // 192MB
  float* t   = (float*)(ws + off); off += (size_t)BATCH * IDIM * SDIM * 4; // 64MB
  float* xn1 = (float*)(ws + off); off += (size_t)BATCH * IDIM * SDIM * 4; // 64MB
  float* w1t = (float*)(ws + off); off += (size_t)6144 * CPG * KW * 4;     // 88MB
  float* x2  = (float*)(ws + off); off += (size_t)BATCH * 6144 * SDIM * 4; // 192MB
  float* xn2 = (float*)(ws + off); off += (size_t)BATCH * IDIM * SDIM * 4; // 64MB

  // 1) x1 = w0 x inp   [6144,1024] x [b][1024,S]
  gemm_f32_wmma<<<dim3(SDIM / 128, 6144 / 128, BATCH), 256, 0, stream>>>(
      w0, inp, x1, 6144, FDIM, SDIM);

  // 2) cumsum/divisor*scale + shift
  scan_kernel<<<(BATCH * IDIM * 32) / 256, 256, 0, stream>>>(x1, divisor, t, SDIM);

  // 3) channel norm + leaky
  norm_kernel<false><<<dim3(SDIM / 32, BATCH), 256, 0, stream>>>(t, xn1, SDIM);

  // 4) transpose w1 for coalesced conv A-tiles
  {
    int total = 6144 * CPG * KW;
    w1_transpose<<<(total + 255) / 256, 256, 0, stream>>>(w1, w1t, total);
  }

  // 5) grouped causal conv via WMMA
  gconv_f32_wmma<<<dim3(SDIM / 128, 48, BATCH), 256, 0, stream>>>(w1t, xn1, x2, SDIM);

  // 6) fused s0*s1+sh -> channel norm + leaky
  norm_kernel<true><<<dim3(SDIM / 32, BATCH), 256, 0, stream>>>(x2, xn2, SDIM);

  // 7) out = w2 x xn2   [1024,2048] x [b][2048,S]
  gemm_f32_wmma<<<dim3(SDIM / 128, FDIM / 128, BATCH), 256, 0, stream>>>(
      w2, xn2, out, FDIM, IDIM, SDIM);
}